// SparseRegistration_27281632264791
// MI455X (gfx1250) — compile-verified
//
#include <hip/hip_runtime.h>
#include <math.h>

// Reference constants (fixed in the reference file)
#define VOL    256
#define IMG_W  512
#define PATCH  13
#define RAD    6            // PATCH // 2
#define PRAD   7            // RAD + 1
#define FW     (IMG_W - 2 * PRAD)   // 498
#define PPP    (PATCH * PATCH)      // 169

typedef float v2f __attribute__((ext_vector_type(2)));
typedef float v8f __attribute__((ext_vector_type(8)));

// ---------------------------------------------------------------------------
// Kernel 1: pose setup. Computes R (Euler ZYX), src' = R@src + t. 1 thread.
// hdr layout (floats): [0..8]=R row-major, [9..11]=src', [12..14]=t, [15]=pad
// ---------------------------------------------------------------------------
__global__ __launch_bounds__(32) void pose_setup_kernel(
    const float* __restrict__ rot, const float* __restrict__ trans,
    const float* __restrict__ det_source, float* __restrict__ hdr)
{
    if (threadIdx.x != 0 || blockIdx.x != 0) return;
    float a = rot[0], b = rot[1], g = rot[2];
    float ca = cosf(a), sa = sinf(a);
    float cb = cosf(b), sb = sinf(b);
    float cg = cosf(g), sg = sinf(g);
    float R[9];
    R[0] = ca * cb; R[1] = ca * sb * sg - sa * cg; R[2] = ca * sb * cg + sa * sg;
    R[3] = sa * cb; R[4] = sa * sb * sg + ca * cg; R[5] = sa * sb * cg - ca * sg;
    R[6] = -sb;     R[7] = cb * sg;                R[8] = cb * cg;
    float sx = det_source[0], sy = det_source[1], sz = det_source[2];
    float tx = trans[0], ty = trans[1], tz = trans[2];
#pragma unroll
    for (int k = 0; k < 9; ++k) hdr[k] = R[k];
    hdr[9]  = R[0] * sx + R[1] * sy + R[2] * sz + tx;
    hdr[10] = R[3] * sx + R[4] * sy + R[5] * sz + ty;
    hdr[11] = R[6] * sx + R[7] * sy + R[8] * sz + tz;
    hdr[12] = tx; hdr[13] = ty; hdr[14] = tz;
    hdr[15] = 0.0f;
}

// Map ray id -> flat detector pixel id (matches reference patch indexing)
__device__ __forceinline__ int ray_pixel(int p, const int* __restrict__ pidx)
{
    int patch = p / PPP;
    int cell  = p - patch * PPP;
    int ci = pidx[patch];
    int i  = ci / FW + PRAD;
    int j  = ci - (ci / FW) * FW + PRAD;
    int di = cell / PATCH - RAD;
    int dj = cell - (cell / PATCH) * PATCH - RAD;
    return (i + di) * IMG_W + (j + dj);
}

// ---------------------------------------------------------------------------
// Kernel 2: pose-transform 16 target points per wave with one
// V_WMMA_F32_16X16X4_F32:  D(16x16) = A(16x4) * B(4x16) + 0
//   A rows 0..2 = [R[m][0], R[m][1], R[m][2], t[m]], rows 3..15 = 0
//   B cols n    = [x_n; y_n; z_n; 1]
//   D rows 0..2 = transformed x/y/z of the 16 points.
// VGPR layouts per CDNA5 ISA 7.12.2:
//   A: v0 = {K0 | K2}, v1 = {K1 | K3} across lane halves, M = lane&15
//   B: v0 = {row0 | row2}, v1 = {row1 | row3}, N = lane&15
//   D: VGPR m, lanes 0-15 = (M=m, N=lane)
// EXEC must be all ones: no divergence before the WMMA (tail ids clamped).
// ---------------------------------------------------------------------------
__global__ __launch_bounds__(32) void transform_targets_wmma(
    const float* __restrict__ det_target, const int* __restrict__ pidx,
    const float* __restrict__ hdr, float* __restrict__ tgt_out, int n_rays)
{
    const int lane = threadIdx.x;          // 0..31
    const int m    = lane & 15;            // point slot / matrix row
    int p  = blockIdx.x * 16 + m;          // ray id handled by this slot
    int pc = p < n_rays ? p : (n_rays - 1);
    int pix = ray_pixel(pc, pidx);

    // A matrix operand: homogeneous [R | t]
    v2f a; a.x = 0.0f; a.y = 0.0f;
    if (m < 3) {
        if (lane < 16) { a.x = hdr[3 * m + 0]; a.y = hdr[3 * m + 1]; } // K=0,1
        else           { a.x = hdr[3 * m + 2]; a.y = hdr[12 + m];    } // K=2,3
    }
    // B matrix operand: point coordinates as columns
    v2f b;
    if (lane < 16) { b.x = det_target[3 * pix + 0]; b.y = det_target[3 * pix + 1]; }
    else           { b.x = det_target[3 * pix + 2]; b.y = 1.0f; }

    v8f c = {};
    c = __builtin_amdgcn_wmma_f32_16x16x4_f32(
            /*neg_a=*/false, a, /*neg_b=*/false, b,
            /*c_mod=*/(short)0, c, /*reuse_a=*/false, /*reuse_b=*/false);

    if (lane < 16 && p < n_rays) {
        tgt_out[3 * p + 0] = c[0];   // row M=0 : x'
        tgt_out[3 * p + 1] = c[1];   // row M=1 : y'
        tgt_out[3 * p + 2] = c[2];   // row M=2 : z'
    }
}

// ---------------------------------------------------------------------------
// Kernel 3: incremental Siddon raycast, one ray per lane.
// ---------------------------------------------------------------------------
__device__ __forceinline__ int clampi(int v, int lo, int hi)
{
    return v < lo ? lo : (v > hi ? hi : v);
}

__global__ __launch_bounds__(256) void siddon_kernel(
    const float* __restrict__ density, const float* __restrict__ spacing,
    const float* __restrict__ hdr, const float* __restrict__ tgt_pre,
    const float* __restrict__ det_target, const int* __restrict__ pidx,
    float* __restrict__ out, int n_rays, int use_pre)
{
    int r = blockIdx.x * blockDim.x + threadIdx.x;
    if (r >= n_rays) return;

    const float s0 = hdr[9], s1 = hdr[10], s2 = hdr[11];
    float t0, t1, t2;
    if (use_pre) {
        t0 = tgt_pre[3 * r + 0];
        t1 = tgt_pre[3 * r + 1];
        t2 = tgt_pre[3 * r + 2];
    } else {
        int pix = ray_pixel(r, pidx);
        float x = det_target[3 * pix + 0];
        float y = det_target[3 * pix + 1];
        float z = det_target[3 * pix + 2];
        t0 = hdr[0] * x + hdr[1] * y + hdr[2] * z + hdr[12];
        t1 = hdr[3] * x + hdr[4] * y + hdr[5] * z + hdr[13];
        t2 = hdr[6] * x + hdr[7] * y + hdr[8] * z + hdr[14];
    }

    const float eps = 1e-8f;
    const float d0 = t0 - s0 + eps;
    const float d1 = t1 - s1 + eps;
    const float d2 = t2 - s2 + eps;
    const float sp0 = spacing[0], sp1 = spacing[1], sp2 = spacing[2];
    const float isp0 = 1.0f / sp0, isp1 = 1.0f / sp1, isp2 = 1.0f / sp2;
    const float id0 = 1.0f / d0, id1 = 1.0f / d1, id2 = 1.0f / d2;
    const float e0 = (float)VOL * sp0, e1 = (float)VOL * sp1, e2 = (float)VOL * sp2;

    // Entry / exit alphas
    float b00 = -s0 * id0, b01 = (e0 - s0) * id0;
    float b10 = -s1 * id1, b11 = (e1 - s1) * id1;
    float b20 = -s2 * id2, b21 = (e2 - s2) * id2;
    float amin = fmaxf(fmaxf(fminf(b00, b01), fminf(b10, b11)),
                       fmaxf(fminf(b20, b21), 0.0f));
    float amax = fminf(fminf(fmaxf(b00, b01), fmaxf(b10, b11)),
                       fminf(fmaxf(b20, b21), 1.0f));
    amax = fmaxf(amax, amin);

    const float BIG = 3.0e38f;
    int   k0, k1, k2, st0, st1, st2;
    float a0n, a1n, a2n;

#define AX_INIT(k, st, an, S, D, ID, SP, ISP)                                   \
    {                                                                           \
        float pm = fmaf(amin, (D), (S)) * (ISP);                                \
        pm = fminf(fmaxf(pm, -2.0f), (float)VOL + 2.0f);                        \
        if ((D) > 0.0f) { st = 1;  k = (int)ceilf(pm); }                        \
        else            { st = -1; k = (int)floorf(pm); }                       \
        an = fmaf((float)k, (SP), -(S)) * (ID);                                 \
        if (an < amin) { k += st; an = fmaf((float)k, (SP), -(S)) * (ID); }     \
        if (k < 0 || k > VOL) an = BIG;                                         \
    }
    AX_INIT(k0, st0, a0n, s0, d0, id0, sp0, isp0)
    AX_INIT(k1, st1, a1n, s1, d1, id1, sp1, isp1)
    AX_INIT(k2, st2, a2n, s2, d2, id2, sp2, isp2)
#undef AX_INIT

    // Dominant-axis voxel stride for lookahead prefetch
    float ad0 = fabsf(d0), ad1 = fabsf(d1), ad2 = fabsf(d2);
    int sdom;
    if (ad0 >= ad1 && ad0 >= ad2) sdom = (d0 > 0.0f ? VOL * VOL : -VOL * VOL);
    else if (ad1 >= ad2)          sdom = (d1 > 0.0f ? VOL : -VOL);
    else                          sdom = (d2 > 0.0f ? 1 : -1);

    float a = amin;
    float acc = 0.0f;

    for (int it = 0; it < 3 * (VOL + 1) + 8; ++it) {
        float an = fminf(fminf(a0n, a1n), a2n);
        if (!(an <= amax)) break;
        float amid = 0.5f * (a + an);
        int ix = clampi((int)floorf(fmaf(amid, d0, s0) * isp0), 0, VOL - 1);
        int iy = clampi((int)floorf(fmaf(amid, d1, s1) * isp1), 0, VOL - 1);
        int iz = clampi((int)floorf(fmaf(amid, d2, s2) * isp2), 0, VOL - 1);
        int addr = (ix * VOL + iy) * VOL + iz;
        float vox = density[addr];
        // Lookahead: next segment's voxel is almost always one step along the
        // dominant axis -> global_prefetch_b8 into WGP$/L2 path.
        int paddr = clampi(addr + sdom, 0, VOL * VOL * VOL - 1);
        __builtin_prefetch(density + paddr, 0, 1);
        acc = fmaf(fmaxf(an - a, 0.0f), vox, acc);
        a = an;
        if (an == a0n) {
            k0 += st0;
            a0n = (k0 >= 0 && k0 <= VOL) ? fmaf((float)k0, sp0, -s0) * id0 : BIG;
        } else if (an == a1n) {
            k1 += st1;
            a1n = (k1 >= 0 && k1 <= VOL) ? fmaf((float)k1, sp1, -s1) * id1 : BIG;
        } else {
            k2 += st2;
            a2n = (k2 >= 0 && k2 <= VOL) ? fmaf((float)k2, sp2, -s2) * id2 : BIG;
        }
    }

    // Tail segment [last_crossing, amax] (reference's first amax-sentinel diff)
    if (amax > a) {
        float amid = 0.5f * (a + amax);
        int ix = clampi((int)floorf(fmaf(amid, d0, s0) * isp0), 0, VOL - 1);
        int iy = clampi((int)floorf(fmaf(amid, d1, s1) * isp1), 0, VOL - 1);
        int iz = clampi((int)floorf(fmaf(amid, d2, s2) * isp2), 0, VOL - 1);
        float vox = density[(ix * VOL + iy) * VOL + iz];
        acc = fmaf(amax - a, vox, acc);
    }

    float raylen = sqrtf(d0 * d0 + d1 * d1 + d2 * d2);
    out[r] = raylen * acc;
}

// ---------------------------------------------------------------------------
// Launcher
// ---------------------------------------------------------------------------
extern "C" void kernel_launch(void* const* d_in, const int* in_sizes, int n_in,
                              void* d_out, int out_size, void* d_ws, size_t ws_size,
                              hipStream_t stream)
{
    (void)n_in; (void)out_size;
    const float* rot   = (const float*)d_in[0];
    const float* trans = (const float*)d_in[1];
    const float* dens  = (const float*)d_in[2];
    const float* spac  = (const float*)d_in[3];
    const float* dsrc  = (const float*)d_in[4];
    const float* dtgt  = (const float*)d_in[5];
    const int*   pidx  = (const int*)d_in[6];

    int n_patches = in_sizes[6];
    int n_rays    = n_patches * PPP;

    float* hdr     = (float*)d_ws;
    float* tgt_pre = hdr + 16;
    int use_pre = (ws_size >= (size_t)(16 + 3 * (size_t)n_rays) * sizeof(float)) ? 1 : 0;

    pose_setup_kernel<<<1, 32, 0, stream>>>(rot, trans, dsrc, hdr);
    if (use_pre) {
        int ngroups = (n_rays + 15) / 16;
        transform_targets_wmma<<<ngroups, 32, 0, stream>>>(dtgt, pidx, hdr, tgt_pre, n_rays);
    }
    int nb = (n_rays + 255) / 256;
    siddon_kernel<<<nb, 256, 0, stream>>>(dens, spac, hdr, tgt_pre, dtgt, pidx,
                                          (float*)d_out, n_rays, use_pre);
}